// MipMap_18382460027615
// MI455X (gfx1250) — compile-verified
//
#include <hip/hip_runtime.h>
#include <hip/hip_bf16.h>

typedef __attribute__((ext_vector_type(2))) float v2f;
typedef __attribute__((ext_vector_type(8))) float v8f;

#define RES   256
#define FEAT  128
#define MIPSZ ((long)RES * RES * FEAT)     // elements per mip level
#define NLVL  6                            // levels used (lvl 0 = base)

// ---------------------------------------------------------------------------
// Separable Gaussian blur pass with 'reflect' boundary (matches
// jnp.pad(mode='reflect') + lax.conv 'same' + crop in the reference).
// One block per (x0, x1) spatial site, 128 threads = feature dim (coalesced).
// ---------------------------------------------------------------------------
__global__ void blur_pass(const float* __restrict__ in, float* __restrict__ out,
                          int M, int axis) {
    __shared__ float w[32];
    __shared__ float s_norm;
    const int f  = threadIdx.x;        // 0..127 feature index
    const int x1 = blockIdx.x;         // fast spatial dim (dim 1)
    const int x0 = blockIdx.y;         // slow spatial dim (dim 0)

    if (f < M) {
        float n = (float)f - 0.5f * (float)(M - 1);   // k - (M-1)/2
        float s = 0.5f * (float)M;                    // std = M/2
        float t = n / s;
        w[f] = __expf(-0.5f * t * t);
    }
    __syncthreads();
    if (f == 0) {
        float t = 0.f;
        for (int k = 0; k < M; ++k) t += w[k];
        s_norm = 1.0f / t;
    }
    __syncthreads();

    const int half = M >> 1;
    const int c    = (axis == 0) ? x0 : x1;
    float acc = 0.f;
    for (int k = 0; k < M; ++k) {
        int j = c + k - half;                 // tap position along `axis`
        j = (j < 0) ? -j : j;                 // reflect (no edge repeat)
        j = (j > RES - 1) ? (2 * (RES - 1) - j) : j;
        long site = (axis == 0) ? ((long)j * RES + x1)
                                : ((long)x0 * RES + j);
        acc += w[k] * in[site * FEAT + f];
    }
    out[((long)x0 * RES + x1) * FEAT + f] = acc * s_norm;
}

// ---------------------------------------------------------------------------
// Fused: 6-level bilinear mip gather (one 16-point tile per block) ->
// layer1 = relu(A@W1+b1) via V_WMMA_F32_16X16X4_F32 (f32 precision) ->
// layer2 = H@W2+b2 (128x4, VALU).
// Block = 128 threads = 4 waves; each wave owns two 16x16 N-tiles of layer 1.
// ---------------------------------------------------------------------------
__global__ void __launch_bounds__(128)
mip_mlp(const float* __restrict__ pt,
        const float* __restrict__ base,
        const float* __restrict__ mips,     // 5 blurred levels in ws
        const float* __restrict__ blv,      // b_levels[6]
        const float* __restrict__ W1,       // 128x128 row-major (K,N)
        const float* __restrict__ b1,       // 128
        const float* __restrict__ W2,       // 128x4
        const float* __restrict__ b2,       // 4
        float* __restrict__ out) {
    __shared__ int   offs[16][4];           // 4 corner element-offsets / point
    __shared__ float wts [16][4];           // 4 bilinear weights / point
    __shared__ float blvS[8];
    __shared__ float fusedL[16][FEAT];      // A matrix (16 x 128), f32
    __shared__ float H[16][FEAT];           // hidden activations

    const int tid = threadIdx.x;
    const int p0  = blockIdx.x * 16;

    if (tid < NLVL) blvS[tid] = blv[tid];
    if (tid < 16) {
        int   p  = p0 + tid;
        // alpha = (pt+1)/2 ; idx_f = alpha*(R-1) ; idx = trunc ; frac = idx_f-idx
        float a0 = (pt[(long)p * 2 + 0] + 1.0f) * 0.5f * (float)(RES - 1);
        float a1 = (pt[(long)p * 2 + 1] + 1.0f) * 0.5f * (float)(RES - 1);
        int i0 = (int)a0;  int i1 = (int)a1;
        i0 = max(0, min(i0, RES - 1));
        i1 = max(0, min(i1, RES - 1));
        float f0 = a0 - (float)i0;
        float f1 = a1 - (float)i1;
        int i0p = min(i0 + 1, RES - 1);
        int i1p = min(i1 + 1, RES - 1);
        // corner index c = c0*2 + c1
        offs[tid][0] = (i0  * RES + i1 ) * FEAT;
        offs[tid][1] = (i0  * RES + i1p) * FEAT;
        offs[tid][2] = (i0p * RES + i1 ) * FEAT;
        offs[tid][3] = (i0p * RES + i1p) * FEAT;
        wts[tid][0] = (1.f - f0) * (1.f - f1);
        wts[tid][1] = (1.f - f0) * f1;
        wts[tid][2] = f0 * (1.f - f1);
        wts[tid][3] = f0 * f1;
    }
    __syncthreads();

    // ---- gather: thread = feature lane, loop points & levels (coalesced 512B rows)
    {
        const int f = tid;   // 0..127
        for (int p = 0; p < 16; ++p) {
            const int   o0 = offs[p][0], o1 = offs[p][1], o2 = offs[p][2], o3 = offs[p][3];
            const float w0 = wts[p][0],  w1 = wts[p][1],  w2 = wts[p][2],  w3 = wts[p][3];
            float acc = 0.f;
#pragma unroll
            for (int lvl = 0; lvl < NLVL; ++lvl) {
                const float* __restrict__ mp =
                    (lvl == 0) ? base : (mips + (long)(lvl - 1) * MIPSZ);
                float v = w0 * mp[o0 + f] + w1 * mp[o1 + f]
                        + w2 * mp[o2 + f] + w3 * mp[o3 + f];
                acc += blvS[lvl] * v;
            }
            fusedL[p][f] = acc;
        }
    }
    __syncthreads();

    // ---- layer 1: D(16x128) = A(16x128) @ W1(128x128) with f32 WMMA, K-step 4
    {
        const int lane = tid & 31;
        const int wave = tid >> 5;
        const int m    = lane & 15;      // row (A) / col-in-tile (B,D)
        const int hi   = lane >> 4;      // K half-select: lanes16-31 hold K={2,3}
#pragma unroll
        for (int t = 0; t < 2; ++t) {
            const int n0  = (wave * 2 + t) * 16;
            const int col = n0 + m;
            v8f acc = {};
            for (int kk = 0; kk < FEAT; kk += 4) {
                // A 16x4 layout: lane<16 -> K={0,1}; lane>=16 -> K={2,3}
                v2f a = *(const v2f*)&fusedL[m][kk + 2 * hi];
                // B 4x16 layout mirrors A with lane = N
                v2f b;
                b.x = W1[(long)(kk + 2 * hi)     * FEAT + col];
                b.y = W1[(long)(kk + 2 * hi + 1) * FEAT + col];
                acc = __builtin_amdgcn_wmma_f32_16x16x4_f32(
                        false, a, false, b, (short)0, acc, false, false);
            }
            const float bias = b1[col];
#pragma unroll
            for (int r = 0; r < 8; ++r) {
                // D layout: VGPR r -> row r (lanes 0-15) / row r+8 (lanes 16-31)
                float h = acc[r] + bias;
                H[r + 8 * hi][col] = h > 0.f ? h : 0.f;
            }
        }
    }
    __syncthreads();

    // ---- layer 2: (16x128) @ (128x4) + b2 -> 64 outputs
    if (tid < 64) {
        const int p = tid >> 2;
        const int j = tid & 3;
        float acc = b2[j];
        for (int k = 0; k < FEAT; ++k)
            acc += H[p][k] * W2[k * 4 + j];
        out[(long)(p0 + p) * 4 + j] = acc;
    }
}

// ---------------------------------------------------------------------------
extern "C" void kernel_launch(void* const* d_in, const int* in_sizes, int n_in,
                              void* d_out, int out_size, void* d_ws, size_t ws_size,
                              hipStream_t stream) {
    const float* pt   = (const float*)d_in[0];   // (N,2)
    const float* base = (const float*)d_in[1];   // (256,256,128)
    const float* blv  = (const float*)d_in[2];   // (6,)
    const float* W1   = (const float*)d_in[3];   // (128,128)
    const float* b1   = (const float*)d_in[4];   // (128,)
    const float* W2   = (const float*)d_in[5];   // (128,4)
    const float* b2   = (const float*)d_in[6];   // (4,)
    float*       out  = (float*)d_out;

    float* mips = (float*)d_ws;                  // 5 mips, 33.5MB each
    float* tmp  = mips + 5 * MIPSZ;              // x-pass scratch

    dim3 bgrid(RES, RES);
    for (int lvl = 1; lvl <= 5; ++lvl) {
        int M = 1 << lvl;
        float* dst = mips + (long)(lvl - 1) * MIPSZ;
        blur_pass<<<bgrid, 128, 0, stream>>>(base, tmp, M, 0);  // along dim 0
        blur_pass<<<bgrid, 128, 0, stream>>>(tmp,  dst, M, 1);  // along dim 1
    }

    const int npts = in_sizes[0] / 2;
    mip_mlp<<<npts / 16, 128, 0, stream>>>(pt, base, mips, blv,
                                           W1, b1, W2, b2, out);
}